// Unsigned_MSELoss_10582799417863
// MI455X (gfx1250) — compile-verified
//
#include <hip/hip_runtime.h>

typedef __attribute__((ext_vector_type(2))) float v2f;
typedef __attribute__((ext_vector_type(8))) float v8f;

#define MAXB 32
#define WAVES_PER_BLOCK 8
#define BLOCK (WAVES_PER_BLOCK * 32)

// ws layout: ws[0] = sum(x^2)+sum(t^2); ws[1..MAXB] = dots[g]
__global__ void usmse_init(float* ws) {
    int i = threadIdx.x;
    if (i <= MAXB) ws[i] = 0.0f;
}

// Diagonal of the 16x16 f32 C/D WMMA layout:
//   lanes 0-7   hold D[L][L]        in c[lane]
//   lanes 24-31 hold D[L-16][L-16]  in c[lane-24]
__device__ __forceinline__ float diag_extract(v8f c, int lane) {
    const int sel = lane & 7;
    float d = c[0];
    d = (sel == 1) ? c[1] : d;
    d = (sel == 2) ? c[2] : d;
    d = (sel == 3) ? c[3] : d;
    d = (sel == 4) ? c[4] : d;
    d = (sel == 5) ? c[5] : d;
    d = (sel == 6) ? c[6] : d;
    d = (sel == 7) ? c[7] : d;
    return d;
}

__global__ __launch_bounds__(BLOCK) void usmse_main(const float* __restrict__ x,
                                                    const float* __restrict__ t,
                                                    const int*   __restrict__ batch,
                                                    int n, float* __restrict__ ws) {
    __shared__ float sdots[MAXB];
    __shared__ float sbase[WAVES_PER_BLOCK];
    for (int i = threadIdx.x; i < MAXB; i += BLOCK) sdots[i] = 0.0f;
    __syncthreads();

    const int  lane  = threadIdx.x & 31;              // wave32
    const int  wib   = threadIdx.x >> 5;
    const int  gwave = blockIdx.x * WAVES_PER_BLOCK + wib;   // unique in [0, nwav)
    const int  nwav  = gridDim.x * WAVES_PER_BLOCK;

    const int  rowInTile = lane & 15;
    const bool hiHalf    = lane >= 16;
    const int  colOff    = hiHalf ? 2 : 0;   // A 16x4 f32: lanes<16 K=0,1; lanes>=16 K=2,3
    const bool diagLane  = (lane < 8) || (lane >= 24);
    const int  node      = hiHalf ? (lane - 16) : lane;

    const int fullTiles = n >> 4;            // tiles with all 16 rows in-range

    float baseAcc = 0.0f;

    // ---- Bulk loop: unguarded loads -> all 8 b64 loads in flight, then 4 WMMAs ----
    for (int tile = gwave; tile < fullTiles; tile += nwav) {
        const int       row = (tile << 4) + rowInTile;
        const long long rb  = (long long)row << 4;    // row * F (F == 16)
        const float* __restrict__ xp = x + rb + colOff;
        const float* __restrict__ tp = t + rb + colOff;

        // Prefetch the next tile this wave will touch (global_prefetch_b8).
        {
            const int nrow = row + (nwav << 4);
            if (nrow < n) {
                __builtin_prefetch(x + ((long long)nrow << 4), 0, 0);
                __builtin_prefetch(t + ((long long)nrow << 4), 0, 0);
            }
        }

        const v2f a0 = *(const v2f*)(xp);
        const v2f a1 = *(const v2f*)(xp + 4);
        const v2f a2 = *(const v2f*)(xp + 8);
        const v2f a3 = *(const v2f*)(xp + 12);
        const v2f b0 = *(const v2f*)(tp);
        const v2f b1 = *(const v2f*)(tp + 4);
        const v2f b2 = *(const v2f*)(tp + 8);
        const v2f b3 = *(const v2f*)(tp + 12);

        // Every (row,col) element is loaded exactly once across the wave:
        // sum(x^2)+sum(t^2) comes for free from the same registers.
        baseAcc += a0[0]*a0[0] + a0[1]*a0[1] + b0[0]*b0[0] + b0[1]*b0[1];
        baseAcc += a1[0]*a1[0] + a1[1]*a1[1] + b1[0]*b1[0] + b1[1]*b1[1];
        baseAcc += a2[0]*a2[0] + a2[1]*a2[1] + b2[0]*b2[0] + b2[1]*b2[1];
        baseAcc += a3[0]*a3[0] + a3[1]*a3[1] + b3[0]*b3[0] + b3[1]*b3[1];

        // D = X_tile * T_tile^T accumulated over K=16; diag(D) = <x_i, t_i>.
        v8f c = {};
        c = __builtin_amdgcn_wmma_f32_16x16x4_f32(false, a0, false, b0, (short)0, c, false, false);
        c = __builtin_amdgcn_wmma_f32_16x16x4_f32(false, a1, false, b1, (short)0, c, false, false);
        c = __builtin_amdgcn_wmma_f32_16x16x4_f32(false, a2, false, b2, (short)0, c, false, false);
        c = __builtin_amdgcn_wmma_f32_16x16x4_f32(false, a3, false, b3, (short)0, c, false, false);

        const float d = diag_extract(c, lane);
        if (diagLane) {
            const int drow = (tile << 4) + node;      // < fullTiles*16 <= n
            const int g    = batch[drow] & (MAXB - 1);
            atomicAdd(&sdots[g], d);                  // ds_add_f32
        }
    }

    // ---- Tail tile (at most one, rows may be out of range): guarded path ----
    if ((n & 15) && (fullTiles % nwav) == gwave) {
        const int       tile  = fullTiles;
        const int       row   = (tile << 4) + rowInTile;
        const bool      rowOk = row < n;
        const long long rb    = (long long)row << 4;

        v8f c = {};
        #pragma unroll
        for (int s = 0; s < 4; ++s) {
            v2f a = {}, b = {};
            if (rowOk) {
                a = *(const v2f*)(x + rb + s * 4 + colOff);
                b = *(const v2f*)(t + rb + s * 4 + colOff);
            }
            baseAcc += a[0]*a[0] + a[1]*a[1] + b[0]*b[0] + b[1]*b[1];
            c = __builtin_amdgcn_wmma_f32_16x16x4_f32(false, a, false, b,
                                                      (short)0, c, false, false);
        }
        const float d    = diag_extract(c, lane);
        const int   drow = (tile << 4) + node;
        if (diagLane && drow < n) {
            const int g = batch[drow] & (MAXB - 1);
            atomicAdd(&sdots[g], d);
        }
    }

    // ---- Wave reduce baseAcc, block reduce, single global atomic per block ----
    #pragma unroll
    for (int m = 16; m >= 1; m >>= 1)
        baseAcc += __shfl_xor(baseAcc, m, 32);
    if (lane == 0) sbase[wib] = baseAcc;
    __syncthreads();

    if (threadIdx.x == 0) {
        float s = 0.0f;
        #pragma unroll
        for (int w = 0; w < WAVES_PER_BLOCK; ++w) s += sbase[w];
        atomicAdd(&ws[0], s);
    }
    if (threadIdx.x < MAXB)
        atomicAdd(&ws[1 + threadIdx.x], sdots[threadIdx.x]);
}

// loss = base - 2 * sum_g |dots[g]|  (exact closed form of the 2^B sign sweep)
__global__ void usmse_finalize(const float* __restrict__ ws,
                               const int* __restrict__ bsz,
                               float* __restrict__ out) {
    if (threadIdx.x == 0 && blockIdx.x == 0) {
        int B = bsz ? bsz[0] : 8;
        if (B < 1) B = 1;
        if (B > MAXB) B = MAXB;
        float s = 0.0f;
        for (int g = 0; g < B; ++g) s += fabsf(ws[1 + g]);
        out[0] = ws[0] - 2.0f * s;
    }
}

extern "C" void kernel_launch(void* const* d_in, const int* in_sizes, int n_in,
                              void* d_out, int out_size, void* d_ws, size_t ws_size,
                              hipStream_t stream) {
    const float* x     = (const float*)d_in[0];
    const float* t     = (const float*)d_in[1];
    const int*   batch = (const int*)d_in[2];
    const int*   bsz   = (n_in > 3) ? (const int*)d_in[3] : nullptr;
    float*       ws    = (float*)d_ws;
    const int    n     = in_sizes[2];            // node count (batch length)

    usmse_init<<<1, 64, 0, stream>>>(ws);

    const int ntile = (n + 15) / 16;
    int blocks = (ntile + WAVES_PER_BLOCK - 1) / WAVES_PER_BLOCK;
    if (blocks > 4096) blocks = 4096;            // ~32K waves: enough to saturate 23.3 TB/s
    if (blocks < 1)    blocks = 1;
    usmse_main<<<blocks, BLOCK, 0, stream>>>(x, t, batch, n, ws);

    usmse_finalize<<<1, 32, 0, stream>>>(ws, bsz, (float*)d_out);
}